// AssociatorCurrent_51410758533303
// MI455X (gfx1250) — compile-verified
//
#include <hip/hip_runtime.h>

typedef float v2f __attribute__((ext_vector_type(2)));
typedef float v8f __attribute__((ext_vector_type(8)));

#define TOK_TOTAL 16384
#define DM 1024
#define BLK_THREADS 128
#define TOK_PER_WAVE 32
#define TOK_PER_BLK 128
#define KC 32
#define NCHUNK (DM / KC)

// LDS layout (floats):
//   Phase 1: [0,4608)  per-wave x tiles (wave w at w*1152, row stride 36, 32 rows)
//            [4608,6144) W tiles (mat m at 4608 + m*512, 32x16 row-major)
//   Phase 2+ (overlap): [0,6144) psi[token_local][mat][comp], stride 48
//   [6144,8192): Jv[token_local][16]
#define LDS_FLOATS 8192
#define XROW_STRIDE 36
#define XBASE(w) ((w) * 1152)
#define WBASE(m) (4608 + (m) * 512)
#define PSI(t, m, c) ((t) * 48 + (m) * 16 + (c))
#define JVBASE 6144

__device__ __forceinline__ void oct_mul8(const float* a, const float* b, float* o) {
#pragma unroll
  for (int i = 0; i < 8; ++i) o[i] = 0.0f;
  constexpr int TR[7][3] = {{0,1,2},{0,3,4},{0,5,6},{1,3,5},{1,4,6},{2,3,6},{2,4,5}};
#pragma unroll
  for (int t = 0; t < 7; ++t) {
    const int i = TR[t][0], j = TR[t][1], k = TR[t][2];
    o[k] += a[i] * b[j] - a[j] * b[i];
    o[i] += a[j] * b[k] - a[k] * b[j];
    o[j] += a[k] * b[i] - a[i] * b[k];
  }
}

__device__ __forceinline__ void complex_oct_mul16(const float* x, const float* y, float* o) {
  float ac[8], bd[8], ad[8], bc[8];
  oct_mul8(x,     y,     ac);
  oct_mul8(x + 8, y + 8, bd);
  oct_mul8(x,     y + 8, ad);
  oct_mul8(x + 8, y,     bc);
#pragma unroll
  for (int i = 0; i < 8; ++i) {
    o[i]     = ac[i] - bd[i];
    o[8 + i] = ad[i] + bc[i];
  }
}

// Prologue: antisymmetrize J_expand along its last two axes.
// Ja[i][j][k] = Je[i][j][k] - Je[i][k][j]   (4096 floats into d_ws)
__global__ void asym_prologue(const float* __restrict__ Je, float* __restrict__ Ja) {
  int t = blockIdx.x * blockDim.x + threadIdx.x;
  if (t < 4096) {
    int i = t >> 8, j = (t >> 4) & 15, k = t & 15;
    Ja[t] = Je[i * 256 + j * 16 + k] - Je[i * 256 + k * 16 + j];
  }
}

__global__ __launch_bounds__(BLK_THREADS)
void associator_fused(const float* __restrict__ x,
                      const float* __restrict__ Ws, const float* __restrict__ bs,
                      const float* __restrict__ Wl, const float* __restrict__ bl,
                      const float* __restrict__ Wa, const float* __restrict__ ba,
                      const float* __restrict__ Jasym,
                      float* __restrict__ out) {
  __shared__ float smem[LDS_FLOATS];
  const int tid  = threadIdx.x;
  const int w    = tid >> 5;       // wave in block
  const int lane = tid & 31;
  const int h    = lane >> 4;      // half-wave select
  const int l16  = lane & 15;
  const int wave_tok0 = blockIdx.x * TOK_PER_BLK + w * TOK_PER_WAVE;

  const float* Wmat[3] = {Ws, Wl, Wa};
  const float* bias[3] = {bs, bl, ba};

  v8f acc[3][2];
#pragma unroll
  for (int m = 0; m < 3; ++m)
#pragma unroll
    for (int t = 0; t < 2; ++t) acc[m][t] = v8f{0.f,0.f,0.f,0.f,0.f,0.f,0.f,0.f};

  // ---------------- Phase 1: psi = x @ W  via V_WMMA_F32_16X16X4_F32 ----------
  for (int c = 0; c < NCHUNK; ++c) {
    __syncthreads();
    const int k0 = c * KC;
    // stage x tile (per-wave, coalesced 128B per 8 lanes)
#pragma unroll
    for (int i = 0; i < 8; ++i) {
      int idx = i * 32 + lane;
      int row = idx >> 3;
      int cg  = idx & 7;
      const float4 v = *reinterpret_cast<const float4*>(
          &x[(size_t)(wave_tok0 + row) * DM + k0 + cg * 4]);
      *reinterpret_cast<float4*>(&smem[XBASE(w) + row * XROW_STRIDE + cg * 4]) = v;
    }
    // stage W tiles (block-cooperative: 128 threads x 16B = one 2KB tile each)
#pragma unroll
    for (int m = 0; m < 3; ++m) {
      const float4 v = *reinterpret_cast<const float4*>(&Wmat[m][k0 * 16 + tid * 4]);
      *reinterpret_cast<float4*>(&smem[WBASE(m) + tid * 4]) = v;
    }
    __syncthreads();
#pragma unroll
    for (int kk = 0; kk < 8; ++kk) {
      const int kl = kk * 4 + 2 * h;   // A: VGPR0/1 hold K=kl,kl+1 (lanes 0-15), K=kl+2.. via h
      v2f a0 = *reinterpret_cast<v2f*>(&smem[XBASE(w) + l16 * XROW_STRIDE + kl]);
      v2f a1 = *reinterpret_cast<v2f*>(&smem[XBASE(w) + (16 + l16) * XROW_STRIDE + kl]);
#pragma unroll
      for (int m = 0; m < 3; ++m) {
        v2f bf;
        bf.x = smem[WBASE(m) + kl * 16 + l16];
        bf.y = smem[WBASE(m) + (kl + 1) * 16 + l16];
        acc[m][0] = __builtin_amdgcn_wmma_f32_16x16x4_f32(
            false, a0, false, bf, (short)0, acc[m][0], false, false);
        acc[m][1] = __builtin_amdgcn_wmma_f32_16x16x4_f32(
            false, a1, false, bf, (short)0, acc[m][1], false, false);
      }
    }
  }
  __syncthreads();

  // ---------------- Phase 2: bias add + transpose psi into LDS ----------------
#pragma unroll
  for (int m = 0; m < 3; ++m) {
    const float bv = bias[m][l16];
#pragma unroll
    for (int t = 0; t < 2; ++t)
#pragma unroll
      for (int r = 0; r < 8; ++r) {
        const int tl = w * 32 + t * 16 + 8 * h + r;  // C layout: M = 8*h + r, N = l16
        smem[PSI(tl, m, l16)] = acc[m][t][r] + bv;
      }
  }
  __syncthreads();

  // ---------------- Phase 3: octonion associator, one token per thread --------
  {
    float ps[16], pl[16], pa[16];
#pragma unroll
    for (int i = 0; i < 16; ++i) {
      ps[i] = smem[PSI(tid, 0, i)];
      pl[i] = smem[PSI(tid, 1, i)];
      pa[i] = smem[PSI(tid, 2, i)];
    }
    float prod_sl[16], left[16], prod_la[16], right[16];
    complex_oct_mul16(ps, pl, prod_sl);
    complex_oct_mul16(prod_sl, pa, left);
    complex_oct_mul16(pl, pa, prod_la);
    complex_oct_mul16(ps, prod_la, right);
#pragma unroll
    for (int i = 0; i < 16; ++i)
      smem[JVBASE + tid * 16 + i] = left[i] - right[i];
  }
  __syncthreads();

  // ---------------- Phase 4: out = Jv @ Jasym (16 -> 256) via WMMA ------------
  v2f afrag[2][4];
#pragma unroll
  for (int t = 0; t < 2; ++t)
#pragma unroll
    for (int ks = 0; ks < 4; ++ks) {
      const int row = t * 16 + l16;
      afrag[t][ks] = *reinterpret_cast<v2f*>(
          &smem[JVBASE + row * 16 + ks * 4 + 2 * h]);
    }
#pragma unroll
  for (int n = 0; n < 16; ++n) {
    v2f bfr[4];
#pragma unroll
    for (int ks = 0; ks < 4; ++ks) {
      const int i0 = ks * 4 + 2 * h;
      bfr[ks].x = Jasym[i0 * 256 + n * 16 + l16];
      bfr[ks].y = Jasym[(i0 + 1) * 256 + n * 16 + l16];
    }
#pragma unroll
    for (int t = 0; t < 2; ++t) {
      v8f cacc = v8f{0.f,0.f,0.f,0.f,0.f,0.f,0.f,0.f};
#pragma unroll
      for (int ks = 0; ks < 4; ++ks)
        cacc = __builtin_amdgcn_wmma_f32_16x16x4_f32(
            false, afrag[t][ks], false, bfr[ks], (short)0, cacc, false, false);
#pragma unroll
      for (int r = 0; r < 8; ++r) {
        const int gt = wave_tok0 + t * 16 + 8 * h + r;
        out[(size_t)gt * 256 + n * 16 + l16] = cacc[r];
      }
    }
  }
}

extern "C" void kernel_launch(void* const* d_in, const int* in_sizes, int n_in,
                              void* d_out, int out_size, void* d_ws, size_t ws_size,
                              hipStream_t stream) {
  const float* x  = (const float*)d_in[0];
  const float* Ws = (const float*)d_in[1];
  const float* bs = (const float*)d_in[2];
  const float* Wl = (const float*)d_in[3];
  const float* bl = (const float*)d_in[4];
  const float* Wa = (const float*)d_in[5];
  const float* ba = (const float*)d_in[6];
  const float* Je = (const float*)d_in[7];
  float* out = (float*)d_out;
  float* Ja  = (float*)d_ws;

  asym_prologue<<<16, 256, 0, stream>>>(Je, Ja);
  associator_fused<<<TOK_TOTAL / TOK_PER_BLK, BLK_THREADS, 0, stream>>>(
      x, Ws, bs, Wl, bl, Wa, ba, Ja, out);
}